// MultiLayerGNN_19172734010019
// MI455X (gfx1250) — compile-verified
//
#include <hip/hip_runtime.h>
#include <math.h>

typedef float v2f __attribute__((ext_vector_type(2)));
typedef float v8f __attribute__((ext_vector_type(8)));

// ---------------------------------------------------------------------------
// Degree / normalization kernels
// ---------------------------------------------------------------------------
__global__ void deg_init_kernel(float* deg, int n) {
    int i = blockIdx.x * blockDim.x + threadIdx.x;
    if (i < n) deg[i] = 1.0f;   // self loop contributes 1
}

__global__ void deg_accum_kernel(const long long* __restrict__ dst, float* deg, int e) {
    int i = blockIdx.x * blockDim.x + threadIdx.x;
    if (i < e) atomicAdd(&deg[(int)dst[i]], 1.0f);
}

__global__ void deg_rsqrt_kernel(const float* __restrict__ deg, float* dinv, int n) {
    int i = blockIdx.x * blockDim.x + threadIdx.x;
    if (i < n) dinv[i] = rsqrtf(deg[i]);
}

// ---------------------------------------------------------------------------
// Zero buffer (float4 stores)
// ---------------------------------------------------------------------------
__global__ void zero_kernel(float4* p, size_t n4) {
    size_t i = (size_t)blockIdx.x * blockDim.x + threadIdx.x;
    if (i < n4) p[i] = make_float4(0.f, 0.f, 0.f, 0.f);
}

// ---------------------------------------------------------------------------
// fp32 WMMA GEMM: OUT[M x NC] = A[M x K] * W[K x NC] (+ bias)
// blockDim.x = 32 * (NC/16); each block computes a 16-row strip, wave w does
// columns [16w, 16w+16). Uses V_WMMA_F32_16X16X4_F32, K stepped by 4.
// Requires M % 16 == 0 (no predication: EXEC must stay all-1s for WMMA).
// ---------------------------------------------------------------------------
template <int K, int NC, bool BIAS>
__global__ void gemm_wmma_f32_kernel(const float* __restrict__ A,
                                     const float* __restrict__ W,
                                     const float* __restrict__ bias,
                                     float* __restrict__ out) {
    const int wave  = threadIdx.x >> 5;       // which 16-col tile
    const int lane  = threadIdx.x & 31;
    const int m0    = blockIdx.x * 16;
    const int n0    = wave * 16;
    const int khalf = lane >> 4;              // 0: K={0,1}, 1: K={2,3}
    const int mA    = lane & 15;              // A row for this lane
    const int col   = lane & 15;              // B/D column for this lane

    // A: lane holds row mA, K = k + 2*khalf + {0,1}  -> contiguous float2
    const float* arow = A + (size_t)(m0 + mA) * K + 2 * khalf;
    // B: lane holds column col, K = k + 2*khalf + {0,1} (row-major W[K][NC])
    const float* wcol = W + (size_t)(2 * khalf) * NC + n0 + col;

    v8f c = {};
#pragma unroll
    for (int k = 0; k < K; k += 4) {
        v2f a;
        a.x = arow[k];
        a.y = arow[k + 1];
        v2f b;
        b.x = wcol[(size_t)k * NC];
        b.y = wcol[(size_t)(k + 1) * NC];
        c = __builtin_amdgcn_wmma_f32_16x16x4_f32(
                /*neg_a=*/false, a, /*neg_b=*/false, b,
                /*c_mod=*/(short)0, c, /*reuse_a=*/false, /*reuse_b=*/false);
    }

    // D layout: VGPR r -> M = r + 8*khalf, N = col
    float bv = 0.f;
    if (BIAS) bv = bias[n0 + col];
#pragma unroll
    for (int r = 0; r < 8; ++r) {
        int row = m0 + r + 8 * khalf;
        float v = c[r];
        if (BIAS) v += bv;
        out[(size_t)row * NC + n0 + col] = v;
    }
}

// ---------------------------------------------------------------------------
// Edge scatter: one wave per edge, 4 channels per lane (128 channels).
// agg[dst] += h[src] * dinv[src]*dinv[dst]
// ---------------------------------------------------------------------------
__global__ void scatter_kernel(const float* __restrict__ h,
                               const long long* __restrict__ src,
                               const long long* __restrict__ dst,
                               const float* __restrict__ dinv,
                               float* __restrict__ agg, int e) {
    int wave = (int)((blockIdx.x * (size_t)blockDim.x + threadIdx.x) >> 5);
    int lane = threadIdx.x & 31;
    if (wave >= e) return;
    int s = (int)src[wave];
    int d = (int)dst[wave];
    float norm = dinv[s] * dinv[d];
    const float4 v = *(const float4*)(h + (size_t)s * 128 + lane * 4);
    float* base = agg + (size_t)d * 128 + lane * 4;
    atomicAdd(base + 0, v.x * norm);
    atomicAdd(base + 1, v.y * norm);
    atomicAdd(base + 2, v.z * norm);
    atomicAdd(base + 3, v.w * norm);
}

// ---------------------------------------------------------------------------
// Epilogue: out = relu(agg + h*dinv^2 (self loop) + b). May alias h_in/out.
// ---------------------------------------------------------------------------
__global__ void relu_combine_kernel(const float* __restrict__ agg,
                                    const float* __restrict__ h_in,
                                    const float* __restrict__ dinv,
                                    const float* __restrict__ b,
                                    float* __restrict__ out, int n) {
    size_t idx = (size_t)blockIdx.x * blockDim.x + threadIdx.x;
    if (idx >= (size_t)n * 128) return;
    int i = (int)(idx >> 7);
    int c = (int)(idx & 127);
    float selfn = dinv[i] * dinv[i];
    float v = agg[idx] + h_in[idx] * selfn + b[c];
    out[idx] = fmaxf(v, 0.f);
}

// ---------------------------------------------------------------------------
// Host launcher
// ---------------------------------------------------------------------------
extern "C" void kernel_launch(void* const* d_in, const int* in_sizes, int n_in,
                              void* d_out, int out_size, void* d_ws, size_t ws_size,
                              hipStream_t stream) {
    (void)in_sizes; (void)n_in; (void)out_size; (void)ws_size;

    constexpr int N = 50000, E = 600000, C = 128, CO = 64;

    const float*     x  = (const float*)d_in[0];
    const long long* ei = (const long long*)d_in[1];   // int64 [2, E]
    const float*     W1 = (const float*)d_in[2];
    const float*     b1 = (const float*)d_in[3];
    const float*     W2 = (const float*)d_in[4];
    const float*     b2 = (const float*)d_in[5];
    const float*     Wo = (const float*)d_in[6];
    const float*     bo = (const float*)d_in[7];
    float*           out = (float*)d_out;

    const long long* src = ei;
    const long long* dst = ei + E;

    // Workspace layout (floats): dinv[N] | deg[N] | bufH[N*C] | bufG[N*C]
    float* dinv = (float*)d_ws;
    float* deg  = dinv + N;
    float* bufH = deg + N;
    float* bufG = bufH + (size_t)N * C;

    const int T = 256;

    // --- degree + normalization ---
    deg_init_kernel<<<(N + T - 1) / T, T, 0, stream>>>(deg, N);
    deg_accum_kernel<<<(E + T - 1) / T, T, 0, stream>>>(dst, deg, E);
    deg_rsqrt_kernel<<<(N + T - 1) / T, T, 0, stream>>>(deg, dinv, N);

    const size_t NC4   = (size_t)N * C / 4;           // float4 count
    const int    zgrid = (int)((NC4 + T - 1) / T);
    const int    ggrid = N / 16;                      // 3125 (N divisible by 16)
    const int    sgrid = (E + 7) / 8;                 // 8 waves/block, wave per edge
    const int    rgrid = (int)(((size_t)N * C + T - 1) / T);

    // --- layer 1: h = relu(Ahat * (x@W1) + b1) ---
    gemm_wmma_f32_kernel<C, C, false><<<ggrid, 8 * 32, 0, stream>>>(x, W1, nullptr, bufH);
    zero_kernel<<<zgrid, T, 0, stream>>>((float4*)bufG, NC4);
    scatter_kernel<<<sgrid, T, 0, stream>>>(bufH, src, dst, dinv, bufG, E);
    relu_combine_kernel<<<rgrid, T, 0, stream>>>(bufG, bufH, dinv, b1, bufH, N);

    // --- layer 2: h = relu(Ahat * (h@W2) + b2) ---
    gemm_wmma_f32_kernel<C, C, false><<<ggrid, 8 * 32, 0, stream>>>(bufH, W2, nullptr, bufG);
    zero_kernel<<<zgrid, T, 0, stream>>>((float4*)bufH, NC4);
    scatter_kernel<<<sgrid, T, 0, stream>>>(bufG, src, dst, dinv, bufH, E);
    relu_combine_kernel<<<rgrid, T, 0, stream>>>(bufH, bufG, dinv, b2, bufG, N);

    // --- output layer: out = h @ Wo + bo ---
    gemm_wmma_f32_kernel<C, CO, true><<<ggrid, 4 * 32, 0, stream>>>(bufG, Wo, bo, out);
}